// KernelNet2d_2422361555127
// MI455X (gfx1250) — compile-verified
//
#include <hip/hip_runtime.h>

// Problem geometry (from reference)
#define XY    4096      // 64*64 pixels
#define NUMC  16384     // 128*128 output channels
#define NEG_SLOPE 0.1f

typedef __attribute__((ext_vector_type(2))) float v2f;
typedef __attribute__((ext_vector_type(8))) float v8f;

__device__ __forceinline__ float lrelu(float x) {
  return x >= 0.0f ? x : NEG_SLOPE * x;
}

// ---------------------------------------------------------------------------
// Kernel 1: tiny MLP trunk, 2 -> 16 -> 16 -> 4 per pixel.
// Writes features as K-pair interleaved float2: feat2[p*2+0]={f0,f1},
// feat2[p*2+1]={f2,f3}  (matches the WMMA B-matrix lane/K layout so the
// head kernel does a single b64 load per lane per tile).
// ---------------------------------------------------------------------------
__global__ __launch_bounds__(256) void mlp_trunk(
    const float* __restrict__ pos,
    const float* __restrict__ w1, const float* __restrict__ b1,
    const float* __restrict__ w2, const float* __restrict__ b2,
    const float* __restrict__ w3, const float* __restrict__ b3,
    float2* __restrict__ feat2) {
  int p = blockIdx.x * blockDim.x + threadIdx.x;
  if (p >= XY) return;

  float x0 = pos[p];        // channel 0
  float x1 = pos[XY + p];   // channel 1

  float h1[16];
#pragma unroll
  for (int o = 0; o < 16; ++o)
    h1[o] = lrelu(fmaf(w1[o * 2 + 0], x0, fmaf(w1[o * 2 + 1], x1, b1[o])));

  float h2[16];
#pragma unroll
  for (int o = 0; o < 16; ++o) {
    float s = b2[o];
#pragma unroll
    for (int i = 0; i < 16; ++i) s = fmaf(w2[o * 16 + i], h1[i], s);
    h2[o] = lrelu(s);
  }

  float h3[4];
#pragma unroll
  for (int o = 0; o < 4; ++o) {
    float s = b3[o];
#pragma unroll
    for (int i = 0; i < 16; ++i) s = fmaf(w3[o * 16 + i], h2[i], s);
    h3[o] = lrelu(s);
  }

  feat2[p * 2 + 0] = make_float2(h3[0], h3[1]);
  feat2[p * 2 + 1] = make_float2(h3[2], h3[3]);
}

// ---------------------------------------------------------------------------
// Kernel 2: head GEMM  out[16384 x 4096] = w4[16384 x 4] * F[4 x 4096] + b4
// via V_WMMA_F32_16X16X4_F32.  One wave = 16-channel M-tile x 128-pixel span
// (8 unrolled WMMAs reusing the A operand; bias folded into the accumulator).
// Pure streaming-store bound: 256 MB out vs ~320 KB read.
// ---------------------------------------------------------------------------
__global__ __launch_bounds__(256) void head_gemm(
    const float* __restrict__ w4, const float* __restrict__ b4,
    const float2* __restrict__ feat2, float* __restrict__ out) {
  const int lane = threadIdx.x & 31;
  const int lm   = lane & 15;        // row (A) / col (B) within tile
  const int lg   = lane >> 4;        // half-wave selects K pair {0,1} vs {2,3}

  const int wid   = blockIdx.x * (blockDim.x >> 5) + (threadIdx.x >> 5);
  const int mTile = wid >> 5;        // 0..1023 -> 16 output channels each
  const int nGrp  = wid & 31;        // 0..31   -> 128 pixels each
  const int m0 = mTile * 16;
  const int p0 = nGrp * 128;

  // A-matrix: lane lm holds w4[m0+lm][k], k = {2*lg, 2*lg+1}  (contiguous pair)
  v2f a;
  {
    const float2 av = *(const float2*)(w4 + (size_t)(m0 + lm) * 4 + lg * 2);
    a.x = av.x; a.y = av.y;
  }

  // C-matrix preloaded with bias: VGPR r holds rows M = r (lanes 0-15) and
  // M = 8+r (lanes 16-31), uniform across N, so c[r] = b4[m0 + lg*8 + r].
  v8f cb;
  {
    const float* bp = b4 + m0 + lg * 8;
#pragma unroll
    for (int r = 0; r < 8; ++r) cb[r] = bp[r];
  }

  float* obase = out + (size_t)(m0 + lg * 8) * XY;

#pragma unroll
  for (int it = 0; it < 8; ++it) {
    const int p = p0 + it * 16 + lm;

    // B-matrix: lane lm = pixel column, K pair selected by lg -> one b64 load.
    const float2 bv = feat2[(size_t)p * 2 + lg];
    v2f b; b.x = bv.x; b.y = bv.y;

    // D = A x B + bias   (8 args: neg_a, A, neg_b, B, c_mod, C, reuse_a, reuse_b)
    v8f d = __builtin_amdgcn_wmma_f32_16x16x4_f32(
        false, a, false, b, (short)0, cb, false, false);

    // D layout: VGPR r -> row M = (lg ? 8 : 0) + r, col = p.
    float* op = obase + p;
#pragma unroll
    for (int r = 0; r < 8; ++r) op[(size_t)r * XY] = d[r];
  }
}

extern "C" void kernel_launch(void* const* d_in, const int* in_sizes, int n_in,
                              void* d_out, int out_size, void* d_ws, size_t ws_size,
                              hipStream_t stream) {
  const float* pos = (const float*)d_in[0];
  const float* w1  = (const float*)d_in[1];
  const float* b1  = (const float*)d_in[2];
  const float* w2  = (const float*)d_in[3];
  const float* b2  = (const float*)d_in[4];
  const float* w3  = (const float*)d_in[5];
  const float* b3  = (const float*)d_in[6];
  const float* w4  = (const float*)d_in[7];
  const float* b4  = (const float*)d_in[8];
  float* out    = (float*)d_out;
  float2* feat2 = (float2*)d_ws;   // 4096 pixels * 2 float2 = 64 KB

  mlp_trunk<<<XY / 256, 256, 0, stream>>>(pos, w1, b1, w2, b2, w3, b3, feat2);
  // 1024 M-tiles * 32 pixel groups = 32768 waves = 4096 blocks of 8 waves
  head_gemm<<<4096, 256, 0, stream>>>(w4, b4, feat2, out);
}